// ScaledDotProductAttention_4355096838769
// MI455X (gfx1250) — compile-verified
//
#include <hip/hip_runtime.h>

// Scaled dot-product attention with causal + key-padding additive masks.
// B=8, S=4096, D=64, fp32.  Outputs: context (B,S,D) then weights (B,S,S),
// concatenated flat in d_out.
//
// One workgroup (256 threads = 8 wave32) handles 16 query rows of one batch.
// The full 16 x 4096 logit strip is kept resident in LDS (CDNA5 WGP has
// 320KB LDS) so softmax (max, exp, sum, normalize) needs no global traffic
// beyond the mandatory 512MB weight-matrix write.
// Both GEMMs use v_wmma_f32_16x16x4_f32 (exact fp32).

typedef float v2f __attribute__((ext_vector_type(2)));
typedef float v8f __attribute__((ext_vector_type(8)));

#define SEQ    4096
#define DMODEL 64
#define NBATCH 8
#define QTILE  16
#define PITCH  4100          // LDS row pitch (floats); 4100%64=4 dodges bank conflicts
#define MASKV  (-1.0e15f)    // reference's additive mask constant

__global__ __launch_bounds__(256)
void attn_fused_kernel(const float* __restrict__ Q,
                       const float* __restrict__ K,
                       const float* __restrict__ V,
                       const float* __restrict__ PAD,
                       float* __restrict__ out_ctx,
                       float* __restrict__ out_w)
{
    __shared__ float sc[QTILE][PITCH];   // logit / prob strip: 16 x 4096 (padded)
    __shared__ float s_max[QTILE];
    __shared__ float s_inv[QTILE];
    __shared__ float s_O[QTILE][DMODEL]; // cross-wave P*V accumulator
    __shared__ int   s_flag;             // any fully-masked row in this strip?

    const int bx   = blockIdx.x;
    const int b    = bx >> 8;            // 256 q-tiles per batch
    const int qt   = bx & 255;
    const int q0   = qt * QTILE;

    const int tid    = threadIdx.x;
    const int wave   = tid >> 5;
    const int lane   = tid & 31;
    const int hi     = lane >> 4;        // which 16-lane half
    const int lane16 = lane & 15;

    const float* qb   = Q   + (size_t)b * SEQ * DMODEL;
    const float* kb   = K   + (size_t)b * SEQ * DMODEL;
    const float* vb   = V   + (size_t)b * SEQ * DMODEL;
    const float* padb = PAD + (size_t)b * SEQ;

    // zero the P*V accumulator
    for (int i = tid; i < QTILE * DMODEL; i += 256) ((float*)s_O)[i] = 0.0f;
    if (tid == 0) s_flag = 0;

    // ---------------- preload Q fragments (A operand) ----------------
    // A is 16x4 fp32 per WMMA: lanes 0-15 hold M=0..15 with K={0,1} in
    // {x,y}; lanes 16-31 hold K={2,3}.  16 K-steps cover D=64.
    v2f afrag[16];
    {
        const float* qrow = qb + (size_t)(q0 + lane16) * DMODEL;
        #pragma unroll
        for (int s = 0; s < 16; ++s) {
            int d0 = 4 * s + 2 * hi;
            afrag[s].x = qrow[d0];
            afrag[s].y = qrow[d0 + 1];
        }
    }

    // ---------------- phase 1: logits -> LDS ----------------
    // wave w owns k-tiles t = w, w+8, ...  (wave-uniform => EXEC all ones)
    for (int t = wave; t < SEQ / 16; t += 8) {
        const int k0 = t * 16;
        if (k0 > q0 + (QTILE - 1)) {
            // fully above the diagonal: logit = -1e15*(1 + pad)  (score absorbed,
            // exactly like the reference's fp32 additive-mask arithmetic)
            float l = MASKV + padb[k0 + lane16] * MASKV;
            #pragma unroll
            for (int vv = 0; vv < 8; ++vv)
                sc[vv + 8 * hi][k0 + lane16] = l;
        } else {
            v8f c = {};
            // B operand 4x16: lanes 0-15 rows K={0,1}, lanes 16-31 rows K={2,3};
            // B[kc][n] = Kmat[k0+n][d]  -> contiguous float2 per lane per step.
            const float* krow = kb + (size_t)(k0 + lane16) * DMODEL;
            #pragma unroll
            for (int s = 0; s < 16; ++s) {
                int d0 = 4 * s + 2 * hi;
                v2f bf;
                bf.x = krow[d0];
                bf.y = krow[d0 + 1];
                c = __builtin_amdgcn_wmma_f32_16x16x4_f32(
                        false, afrag[s], false, bf, (short)0, c, false, false);
            }
            // C layout: element (m = vv + 8*hi, n = lane16)
            const float pn   = padb[k0 + lane16];
            const int   ncol = k0 + lane16;
            #pragma unroll
            for (int vv = 0; vv < 8; ++vv) {
                int   mrow = q0 + vv + 8 * hi;
                float l = c[vv] * 0.125f;          // 1/sqrt(64)
                l = l + pn * MASKV;                // pad mask (same add order as ref)
                if (ncol > mrow) l = l + MASKV;    // look-ahead mask
                sc[vv + 8 * hi][ncol] = l;
            }
        }
    }
    __syncthreads();

    // ---------------- phase 2: row max ----------------
    #pragma unroll
    for (int rr = 0; rr < 2; ++rr) {
        const int r = wave * 2 + rr;
        float m = -3.0e15f;
        for (int c0 = lane * 4; c0 < SEQ; c0 += 128) {
            float4 x = *(const float4*)&sc[r][c0];
            m = fmaxf(m, fmaxf(fmaxf(x.x, x.y), fmaxf(x.z, x.w)));
        }
        #pragma unroll
        for (int off = 16; off > 0; off >>= 1)
            m = fmaxf(m, __shfl_xor(m, off, 32));
        if (lane == 0) s_max[r] = m;
    }
    __syncthreads();
    if (tid < QTILE && s_max[tid] <= -0.9e15f) s_flag = 1;  // degenerate row(s)
    __syncthreads();

    // ---------------- phase 3: exp + row sum ----------------
    #pragma unroll
    for (int rr = 0; rr < 2; ++rr) {
        const int r  = wave * 2 + rr;
        const float rm = s_max[r];
        float sum = 0.0f;
        for (int c0 = lane * 4; c0 < SEQ; c0 += 128) {
            float4 x = *(const float4*)&sc[r][c0];
            x.x = expf(x.x - rm); x.y = expf(x.y - rm);
            x.z = expf(x.z - rm); x.w = expf(x.w - rm);
            sum += (x.x + x.y) + (x.z + x.w);
            *(float4*)&sc[r][c0] = x;
        }
        #pragma unroll
        for (int off = 16; off > 0; off >>= 1)
            sum += __shfl_xor(sum, off, 32);
        if (lane == 0) s_inv[r] = 1.0f / sum;   // sum >= 1 always (exp(0) member)
    }
    __syncthreads();

    // ---------------- phase 4: normalize + stream weights to global ----------------
    {
        float* wbase = out_w + (size_t)b * SEQ * SEQ + (size_t)q0 * SEQ;
        for (int i = tid; i < QTILE * (SEQ / 4); i += 256) {
            const int r  = i >> 10;          // 1024 float4 per row
            const int c4 = (i & 1023) << 2;
            const float inv = s_inv[r];
            float4 x = *(const float4*)&sc[r][c4];
            x.x *= inv; x.y *= inv; x.z *= inv; x.w *= inv;
            *(float4*)&sc[r][c4] = x;                        // probs for P*V
            *(float4*)(wbase + (size_t)r * SEQ + c4) = x;    // coalesced 16B stores
        }
    }
    __syncthreads();

    // ---------------- phase 5: context = P * V ----------------
    // Above-diagonal probs are nonzero only for degenerate rows -> include
    // those tiles only when flagged.
    {
        const int ntiles = s_flag ? (SEQ / 16) : (qt + 1);
        v8f o0 = {}, o1 = {}, o2 = {}, o3 = {};
        for (int t = wave; t < ntiles; t += 8) {
            const int k0 = t * 16;
            #pragma unroll
            for (int s = 0; s < 4; ++s) {
                const int kc = k0 + 4 * s;
                // A operand: P[m][kc + {0,1}] / P[m][kc + {2,3}] per lane half
                v2f a;
                a.x = sc[lane16][kc + 2 * hi];
                a.y = sc[lane16][kc + 2 * hi + 1];
                const float* vp = vb + (size_t)(kc + 2 * hi) * DMODEL + lane16;
                #pragma unroll
                for (int td = 0; td < 4; ++td) {
                    v2f bf;
                    bf.x = vp[16 * td];             // V[kc+2*hi   ][16*td + n]
                    bf.y = vp[16 * td + DMODEL];    // V[kc+2*hi+1 ][16*td + n]
                    v8f& o = (td == 0) ? o0 : (td == 1) ? o1 : (td == 2) ? o2 : o3;
                    o = __builtin_amdgcn_wmma_f32_16x16x4_f32(
                            false, a, false, bf, (short)0, o, false, false);
                }
            }
        }
        // cross-wave reduction via LDS float atomics (ds_add_f32)
        #pragma unroll
        for (int td = 0; td < 4; ++td) {
            const v8f& o = (td == 0) ? o0 : (td == 1) ? o1 : (td == 2) ? o2 : o3;
            #pragma unroll
            for (int vv = 0; vv < 8; ++vv)
                atomicAdd(&s_O[vv + 8 * hi][16 * td + lane16], o[vv]);
        }
    }
    __syncthreads();

    // ---------------- phase 6: write context ----------------
    {
        float* cbase = out_ctx + (size_t)b * SEQ * DMODEL + (size_t)q0 * DMODEL;
        // 16*64 floats = 256 float4 -> one per thread
        float4 x = *(const float4*)&((const float*)s_O)[tid * 4];
        *(float4*)(cbase + tid * 4) = x;
    }
}

extern "C" void kernel_launch(void* const* d_in, const int* in_sizes, int n_in,
                              void* d_out, int out_size, void* d_ws, size_t ws_size,
                              hipStream_t stream) {
    const float* q   = (const float*)d_in[0];
    const float* k   = (const float*)d_in[1];
    const float* v   = (const float*)d_in[2];
    const float* pad = (const float*)d_in[3];
    // d_in[4] (look_ahead_mask) is a deterministic triangular mask -> computed
    // analytically in-kernel from indices.
    float* ctx = (float*)d_out;
    float* w   = (float*)d_out + (size_t)NBATCH * SEQ * DMODEL;
    attn_fused_kernel<<<NBATCH * (SEQ / QTILE), 256, 0, stream>>>(q, k, v, pad, ctx, w);
}